// GAT_46763603919526
// MI455X (gfx1250) — compile-verified
//
#include <hip/hip_runtime.h>
#include <hip/hip_bf16.h>

// Problem constants (match reference)
#define NN  524288      // nodes
#define EE  8388608     // edges
#define FF  128         // input features
#define HH  8           // hidden
#define BBG 64          // graphs
#define NPG (NN / BBG)  // nodes per graph

typedef __attribute__((ext_vector_type(2))) float v2f;
typedef __attribute__((ext_vector_type(8))) float v8f;

// ---------------------------------------------------------------- utilities
__global__ void zero_f32(float* __restrict__ p, long n) {
  long i = (long)blockIdx.x * blockDim.x + threadIdx.x;
  if (i < n) p[i] = 0.0f;
}

// deg[dst] += 1 per edge (segment_sum of ones)
__global__ void deg_count(const int* __restrict__ dst, float* __restrict__ deg) {
  int e = blockIdx.x * blockDim.x + threadIdx.x;
  if (e < EE) atomicAdd(&deg[dst[e]], 1.0f);
}

// dis[i] = rsqrt(deg[i] + 1)  (self-loop included)
__global__ void make_dis(const float* __restrict__ deg, float* __restrict__ dis) {
  int i = blockIdx.x * blockDim.x + threadIdx.x;
  if (i < NN) dis[i] = rsqrtf(deg[i] + 1.0f);
}

// ------------------------------------------------- W1 operand pre-swizzle
// B operand layout for V_WMMA_F32_16X16X4_F32 (mirror of documented A 16x4):
//   lane l in [0,16):  col n=l,    holds K = 4c+0, 4c+1
//   lane l in [16,32): col n=l-16, holds K = 4c+2, 4c+3
// Bsw[(c*32 + lane)*2 + w] = W1pad[K = 4c + 2*(lane>>4) + w][n = lane&15]
// W1 is [128,8] row-major; pad cols 8..15 with zero.
__global__ void prep_b(const float* __restrict__ W1, float* __restrict__ Bsw) {
  int t = blockIdx.x * blockDim.x + threadIdx.x;   // 0 .. 32*32*2-1
  if (t >= 32 * 32 * 2) return;
  int c4    = t >> 6;
  int lane  = (t >> 1) & 31;
  int which = t & 1;
  int k = c4 * 4 + ((lane >> 4) << 1) + which;
  int n = lane & 15;
  Bsw[t] = (n < HH) ? W1[k * HH + n] : 0.0f;
}

// ------------------------------------------------- layer-1 GEMM via WMMA
// h[N,8] = x[N,128] @ W1[128,8], exact fp32 via 32 chained
// V_WMMA_F32_16X16X4_F32 ops per 16-row tile. One wave per tile.
__global__ void gemm_x_w1(const float* __restrict__ x,
                          const float* __restrict__ Bsw,
                          float* __restrict__ h) {
  int wave = (int)((blockIdx.x * blockDim.x + threadIdx.x) >> 5);
  int lane = threadIdx.x & 31;
  int row0 = wave * 16;                 // NN/16 waves, exact
  int m    = lane & 15;
  int khalf = (lane >> 4) << 1;         // 0 or 2: which K-pair this lane holds
  const float* arow = x + (size_t)(row0 + m) * FF;

  v8f acc = {};
  #pragma unroll 8
  for (int c4 = 0; c4 < 32; ++c4) {
    v2f a = *(const v2f*)(arow + c4 * 4 + khalf);
    v2f b = *(const v2f*)(Bsw + ((c4 << 5) + lane) * 2);
    acc = __builtin_amdgcn_wmma_f32_16x16x4_f32(
        /*neg_a=*/false, a, /*neg_b=*/false, b,
        /*c_mod=*/(short)0, acc, /*reuse_a=*/false, /*reuse_b=*/false);
  }

  // C/D layout: VGPR j -> lanes 0-15: (M=j,   N=lane)
  //                       lanes 16-31:(M=j+8, N=lane-16)
  int n = lane & 15;
  int mbase = (lane >> 4) << 3;
  if (n < HH) {
    #pragma unroll
    for (int j = 0; j < 8; ++j)
      h[(size_t)(row0 + mbase + j) * HH + n] = acc[j];
  }
}

// ------------------------------------------------- edge aggregation (H=8)
__global__ void agg8(const int* __restrict__ src, const int* __restrict__ dst,
                     const float* __restrict__ dis, const float* __restrict__ h,
                     float* __restrict__ agg) {
  int e = blockIdx.x * blockDim.x + threadIdx.x;
  if (e >= EE) return;
  int s = src[e], d = dst[e];
  float coef = dis[s] * dis[d];
  const float* hs = h + (size_t)s * HH;
  float* ad = agg + (size_t)d * HH;
  #pragma unroll
  for (int j = 0; j < HH; ++j) atomicAdd(ad + j, hs[j] * coef);
}

// out[i,j] = relu(agg[i,j] + h[i,j] * dis[i]^2 + b[j])
__global__ void post8(const float* __restrict__ agg, const float* __restrict__ h,
                      const float* __restrict__ dis, const float* __restrict__ b,
                      float* __restrict__ out) {
  long t = (long)blockIdx.x * blockDim.x + threadIdx.x;
  if (t >= (long)NN * HH) return;
  long i = t >> 3;
  int  j = (int)(t & 7);
  float di = dis[i];
  float v = agg[t] + h[t] * di * di + b[j];
  out[t] = v > 0.0f ? v : 0.0f;
}

// h[i,:] = xin[i,:] @ W (8x8), per-node
__global__ void mm8x8(const float* __restrict__ xin, const float* __restrict__ W,
                      float* __restrict__ h) {
  int i = blockIdx.x * blockDim.x + threadIdx.x;
  if (i >= NN) return;
  float v[HH];
  #pragma unroll
  for (int k = 0; k < HH; ++k) v[k] = xin[(size_t)i * HH + k];
  #pragma unroll
  for (int j = 0; j < HH; ++j) {
    float s = 0.0f;
    #pragma unroll
    for (int k = 0; k < HH; ++k) s += v[k] * W[k * HH + j];
    h[(size_t)i * HH + j] = s;
  }
}

// h3[i] = x2[i,:] @ W3 (8x1)
__global__ void mv8(const float* __restrict__ xin, const float* __restrict__ W3,
                    float* __restrict__ h3) {
  int i = blockIdx.x * blockDim.x + threadIdx.x;
  if (i >= NN) return;
  float s = 0.0f;
  #pragma unroll
  for (int k = 0; k < HH; ++k) s += xin[(size_t)i * HH + k] * W3[k];
  h3[i] = s;
}

// scalar (H=1) edge aggregation
__global__ void agg1(const int* __restrict__ src, const int* __restrict__ dst,
                     const float* __restrict__ dis, const float* __restrict__ h3,
                     float* __restrict__ agg) {
  int e = blockIdx.x * blockDim.x + threadIdx.x;
  if (e >= EE) return;
  int s = src[e], d = dst[e];
  atomicAdd(&agg[d], h3[s] * dis[s] * dis[d]);
}

__global__ void post1(const float* __restrict__ agg, const float* __restrict__ h3,
                      const float* __restrict__ dis, const float* __restrict__ b3,
                      float* __restrict__ out) {
  int i = blockIdx.x * blockDim.x + threadIdx.x;
  if (i >= NN) return;
  float di = dis[i];
  float v = agg[i] + h3[i] * di * di + b3[0];
  out[i] = v > 0.0f ? v : 0.0f;
}

// ------------------------------------------------- MLP head on 64 targets
__global__ void head(const float* __restrict__ x1, const float* __restrict__ x2,
                     const float* __restrict__ x3, const int* __restrict__ tid,
                     const float* __restrict__ Wl1, const float* __restrict__ bl1,
                     const float* __restrict__ Wl2, const float* __restrict__ bl2,
                     float* __restrict__ out) {
  int g = threadIdx.x;
  if (g >= BBG) return;
  long idx = (long)g * NPG + tid[g];
  float f[17];
  #pragma unroll
  for (int j = 0; j < HH; ++j) {
    f[j]      = x1[idx * HH + j];
    f[HH + j] = x2[idx * HH + j];
  }
  f[16] = x3[idx];
  float z[16];
  #pragma unroll
  for (int j = 0; j < 16; ++j) {
    float s = bl1[j];
    #pragma unroll
    for (int k = 0; k < 17; ++k) s += f[k] * Wl1[k * 16 + j];
    z[j] = s > 0.0f ? s : 0.0f;
  }
  #pragma unroll
  for (int o = 0; o < 2; ++o) {
    float s = bl2[o];
    #pragma unroll
    for (int j = 0; j < 16; ++j) s += z[j] * Wl2[j * 2 + o];
    out[g * 2 + o] = s;
  }
}

// ---------------------------------------------------------------- driver
extern "C" void kernel_launch(void* const* d_in, const int* in_sizes, int n_in,
                              void* d_out, int out_size, void* d_ws, size_t ws_size,
                              hipStream_t stream) {
  const float* x   = (const float*)d_in[0];
  const int*   ei  = (const int*)d_in[1];
  const int*   src = ei;
  const int*   dst = ei + EE;
  const int*   tid = (const int*)d_in[2];
  const float* W1  = (const float*)d_in[3];
  const float* b1  = (const float*)d_in[4];
  const float* W2  = (const float*)d_in[5];
  const float* b2  = (const float*)d_in[6];
  const float* W3  = (const float*)d_in[7];
  const float* b3  = (const float*)d_in[8];
  const float* Wl1 = (const float*)d_in[9];
  const float* bl1 = (const float*)d_in[10];
  const float* Wl2 = (const float*)d_in[11];
  const float* bl2 = (const float*)d_in[12];

  // workspace layout (floats): ~34*N + 2048 ≈ 71 MB
  float* ws  = (float*)d_ws;
  float* dis = ws;                        // [N]
  float* h   = dis + (size_t)NN;          // [8N]  (per-layer pre-agg / h3)
  float* agg = h   + (size_t)8 * NN;      // [8N]  (also deg scratch)
  float* x1  = agg + (size_t)8 * NN;      // [8N]
  float* x2  = x1  + (size_t)8 * NN;      // [8N]
  float* x3  = x2  + (size_t)8 * NN;      // [N]
  float* Bsw = x3  + (size_t)NN;          // [2048] swizzled W1 operand

  const int T = 256;
  const int gE  = EE / T;                 // 32768
  const int gN  = NN / T;                 // 2048
  const int g8N = 8 * NN / T;             // 16384

  // degree -> dis
  zero_f32<<<gN, T, 0, stream>>>(agg, (long)NN);
  deg_count<<<gE, T, 0, stream>>>(dst, agg);
  make_dis<<<gN, T, 0, stream>>>(agg, dis);

  // layer 1: WMMA GEMM, then normalized scatter-add, then relu
  prep_b<<<8, T, 0, stream>>>(W1, Bsw);
  gemm_x_w1<<<NN / 16 / 8, T, 0, stream>>>(x, Bsw, h);   // 8 waves/block
  zero_f32<<<g8N, T, 0, stream>>>(agg, (long)8 * NN);
  agg8<<<gE, T, 0, stream>>>(src, dst, dis, h, agg);
  post8<<<g8N, T, 0, stream>>>(agg, h, dis, b1, x1);

  // layer 2
  mm8x8<<<gN, T, 0, stream>>>(x1, W2, h);
  zero_f32<<<g8N, T, 0, stream>>>(agg, (long)8 * NN);
  agg8<<<gE, T, 0, stream>>>(src, dst, dis, h, agg);
  post8<<<g8N, T, 0, stream>>>(agg, h, dis, b2, x2);

  // layer 3 (H=1)
  mv8<<<gN, T, 0, stream>>>(x2, W3, h);
  zero_f32<<<gN, T, 0, stream>>>(agg, (long)NN);
  agg1<<<gE, T, 0, stream>>>(src, dst, dis, h, agg);
  post1<<<gN, T, 0, stream>>>(agg, h, dis, b3, x3);

  // MLP head on gathered targets
  head<<<1, 64, 0, stream>>>(x1, x2, x3, tid, Wl1, bl1, Wl2, bl2, (float*)d_out);
}